// GGNN_40381282517518
// MI455X (gfx1250) — compile-verified
//
#include <hip/hip_runtime.h>

typedef float v2f __attribute__((ext_vector_type(2)));
typedef float v8f __attribute__((ext_vector_type(8)));

#define N_VARS 1024
#define NMSG   8192
#define E1_TOT 57344
#define SDIM   64
#define CHUNK  8192
#define NCHUNK 7
#define KPAD   132

// ---------------------------------------------------------------------------
// Generic fp32 WMMA GEMM: Y = act(X @ W^T + bias), X[M,K], W[N,K], Y[M,N]
// One wave computes one 16x16 tile of Y via V_WMMA_F32_16X16X4_F32.
// A layout (16x4 f32): lanes0-15 M=lane K={0,1}; lanes16-31 M=lane-16 K={2,3}
// B layout (4x16 f32): lanes0-15 N=lane K={0,1}; lanes16-31 N=lane-16 K={2,3}
// D layout (16x16 f32): reg r -> row r + 8*(lane>=16), col = lane&15
// ---------------------------------------------------------------------------
__global__ void wmma_gemm_bias_act(const float* __restrict__ X,
                                   const float* __restrict__ W,
                                   const float* __restrict__ bias,
                                   float* __restrict__ Y,
                                   int M, int N, int K, int relu) {
    int ntiles = N >> 4;
    int total  = (M >> 4) * ntiles;
    int wave   = blockIdx.x * blockDim.y + threadIdx.y;
    if (wave >= total) return;              // wave-uniform: EXEC stays all-ones
    int mt = wave / ntiles;
    int nt = wave - mt * ntiles;
    int lane = threadIdx.x;
    int half = lane >> 4;                   // 0 or 1
    int l15  = lane & 15;
    const float* xrow = X + (size_t)(mt * 16 + l15) * K;
    const float* wrow = W + (size_t)(nt * 16 + l15) * K;
    v8f acc = {};
    #pragma unroll 4
    for (int k = 0; k < K; k += 4) {
        int ko = k + (half << 1);
        v2f a, bf;
        a.x  = xrow[ko];     a.y  = xrow[ko + 1];
        bf.x = wrow[ko];     bf.y = wrow[ko + 1];
        acc = __builtin_amdgcn_wmma_f32_16x16x4_f32(false, a, false, bf,
                                                    (short)0, acc, false, false);
    }
    int col = nt * 16 + l15;
    float bv = bias[col];
    int rbase = mt * 16 + (half << 3);
    #pragma unroll
    for (int r = 0; r < 8; ++r) {
        float v = acc[r] + bv;
        if (relu) v = fmaxf(v, 0.0f);
        Y[(size_t)(rbase + r) * N + col] = v;
    }
}

// raw1[e] = [ h_f2v[e1_row[e]] (64) | h_v2f[e1_col[e]] (64) | b[var(e1_col[e])] | 0,0,0 ]
__global__ void gather_v2f_raw(float* __restrict__ raw,
                               const float* __restrict__ hf,
                               const float* __restrict__ hv,
                               const int* __restrict__ e1_row,
                               const int* __restrict__ e1_col,
                               const int* __restrict__ node_var,
                               const float* __restrict__ b, int ebase) {
    int idx = blockIdx.x * blockDim.x + threadIdx.x;
    if (idx >= CHUNK * KPAD) return;
    int el = idx / KPAD;
    int c  = idx - el * KPAD;
    int e  = ebase + el;
    float v;
    if (c < 64)        v = hf[e1_row[e] * SDIM + c];
    else if (c < 128)  v = hv[e1_col[e] * SDIM + (c - 64)];
    else if (c == 128) v = b[node_var[e1_col[e]]];
    else               v = 0.0f;
    raw[idx] = v;
}

// raw2[m] = [ h_v2f[m] (64) | h_f2v[m^1] (64) | b[u] | b[v] | 0,0 ]
__global__ void gather_f2v_raw(float* __restrict__ raw,
                               const float* __restrict__ hv,
                               const float* __restrict__ hf,
                               const int* __restrict__ node_var,
                               const float* __restrict__ b) {
    int idx = blockIdx.x * blockDim.x + threadIdx.x;
    if (idx >= NMSG * KPAD) return;
    int m = idx / KPAD;
    int c = idx - m * KPAD;
    float v;
    if (c < 64)        v = hv[m * SDIM + c];
    else if (c < 128)  v = hf[(m ^ 1) * SDIM + (c - 64)];
    else if (c == 128) v = b[node_var[m]];
    else if (c == 129) v = b[node_var[m ^ 1]];
    else               v = 0.0f;
    raw[idx] = v;
}

// em[r] = H[r] @ W3^T + b3   (W3 is [2,128])
__global__ void em_head(const float* __restrict__ H, const float* __restrict__ W3,
                        const float* __restrict__ b3, float* __restrict__ em,
                        int rows, int outBase) {
    int r = blockIdx.x * blockDim.x + threadIdx.x;
    if (r >= rows) return;
    const float* h = H + (size_t)r * 128;
    float a0 = b3[0], a1 = b3[1];
    for (int t = 0; t < 128; ++t) { float hv = h[t]; a0 += hv * W3[t]; a1 += hv * W3[128 + t]; }
    em[(size_t)(outBase + r) * 2 + 0] = a0;
    em[(size_t)(outBase + r) * 2 + 1] = a1;
}

// Deterministic segment-sum replacement: msg node m (var n, rank c) sums the
// 7 meta-edges (a,c), a!=c, located at n*56 + a*7 + (c<a ? c : c-1).
__global__ void nm_gather(const float* __restrict__ em,
                          const int* __restrict__ node_var,
                          const int* __restrict__ inv_c,
                          float* __restrict__ nm) {
    int m = blockIdx.x * blockDim.x + threadIdx.x;
    if (m >= NMSG) return;
    int n = node_var[m], c = inv_c[m];
    float a0 = 0.0f, a1 = 0.0f;
    for (int a = 0; a < 8; ++a) {
        if (a == c) continue;
        int e = n * 56 + a * 7 + (c < a ? c : c - 1);
        a0 += em[e * 2];
        a1 += em[e * 2 + 1];
    }
    nm[m * 2] = a0; nm[m * 2 + 1] = a1;
}

__device__ __forceinline__ float lse2(float a, float b) {
    float mx = fmaxf(a, b);
    return mx + logf(expf(a - mx) + expf(b - mx));
}

// new[m] = logsumexp over factor table [[A+B,-2A],[-2B,A+B]] + em broadcast
__global__ void lse_factor(const float* __restrict__ em2, const float* __restrict__ J,
                           const int* __restrict__ node_var, float* __restrict__ nw) {
    int m = blockIdx.x * blockDim.x + threadIdx.x;
    if (m >= NMSG) return;
    int u = node_var[m], v = node_var[m ^ 1];
    float A = J[(size_t)u * N_VARS + v];
    float B = J[(size_t)v * N_VARS + u];
    float e0 = em2[m * 2], e1 = em2[m * 2 + 1];
    nw[m * 2]     = lse2(A + B + e0, -2.0f * B + e1);
    nw[m * 2 + 1] = lse2(-2.0f * A + e0, A + B + e1);
}

// torch GRUCell (gate order r,z,n); thread = (row, s)
__global__ void gru_update(const float* __restrict__ x, const float* __restrict__ hold,
                           float* __restrict__ hnew, const float* __restrict__ Wih,
                           const float* __restrict__ Whh, const float* __restrict__ bih,
                           const float* __restrict__ bhh, int rows, int sibx) {
    int idx = blockIdx.x * blockDim.x + threadIdx.x;
    if (idx >= rows * SDIM) return;
    int row = idx >> 6, s = idx & 63;
    int xr = sibx ? (row ^ 1) : row;
    float x0 = x[xr * 2], x1 = x[xr * 2 + 1];
    float ir = bih[s]       + Wih[s * 2] * x0         + Wih[s * 2 + 1] * x1;
    float iz = bih[64 + s]  + Wih[(64 + s) * 2] * x0  + Wih[(64 + s) * 2 + 1] * x1;
    float ig = bih[128 + s] + Wih[(128 + s) * 2] * x0 + Wih[(128 + s) * 2 + 1] * x1;
    float hr = bhh[s], hz = bhh[64 + s], hg = bhh[128 + s];
    const float* h  = hold + (size_t)row * SDIM;
    const float* w0 = Whh + (size_t)s * SDIM;
    const float* w1 = Whh + (size_t)(64 + s) * SDIM;
    const float* w2 = Whh + (size_t)(128 + s) * SDIM;
    for (int t = 0; t < SDIM; ++t) {
        float hv = h[t];
        hr += w0[t] * hv; hz += w1[t] * hv; hg += w2[t] * hv;
    }
    float r = 1.0f / (1.0f + expf(-(ir + hr)));
    float z = 1.0f / (1.0f + expf(-(iz + hz)));
    float g = tanhf(ig + r * hg);
    hnew[(size_t)row * SDIM + s] = (1.0f - z) * g + z * h[s];
}

// Pad W1s [128,129]/[128,130] to K=132 with zeros
__global__ void pad_w1(const float* __restrict__ wv, const float* __restrict__ wf,
                       float* __restrict__ pv, float* __restrict__ pf) {
    int idx = blockIdx.x * blockDim.x + threadIdx.x;
    if (idx >= 128 * KPAD) return;
    int n = idx / KPAD, k = idx - n * KPAD;
    pv[idx] = (k < 129) ? wv[n * 129 + k] : 0.0f;
    pf[idx] = (k < 130) ? wf[n * 130 + k] : 0.0f;
}

// order[u][k] = k-th msg node of var u (ascending m = stable argsort);
// inv_c[m] = rank of m within its variable group
__global__ void build_order(const int* __restrict__ node_var,
                            int* __restrict__ order, int* __restrict__ inv_c) {
    int u = blockIdx.x * blockDim.x + threadIdx.x;
    if (u >= N_VARS) return;
    int cnt = 0;
    for (int m = 0; m < NMSG && cnt < 8; ++m) {
        if (node_var[m] == u) { order[u * 8 + cnt] = m; inv_c[m] = cnt; ++cnt; }
    }
}

__global__ void node_msg_sum(const float* __restrict__ hf, const int* __restrict__ order,
                             float* __restrict__ nmsg) {
    int idx = blockIdx.x * blockDim.x + threadIdx.x;
    if (idx >= N_VARS * SDIM) return;
    int u = idx >> 6, s = idx & 63;
    float acc = 0.0f;
    for (int k = 0; k < 8; ++k) acc += hf[order[u * 8 + k] * SDIM + s];
    nmsg[idx] = acc;
}

__global__ void readout_softmax(const float* __restrict__ L2, const float* __restrict__ W3,
                                const float* __restrict__ b3, float* __restrict__ out) {
    int u = blockIdx.x * blockDim.x + threadIdx.x;
    if (u >= N_VARS) return;
    const float* h = L2 + (size_t)u * 128;
    float l0 = b3[0], l1 = b3[1];
    for (int t = 0; t < 128; ++t) { float hv = h[t]; l0 += hv * W3[t]; l1 += hv * W3[128 + t]; }
    float mx = fmaxf(l0, l1);
    float p0 = expf(l0 - mx), p1 = expf(l1 - mx);
    float inv = 1.0f / (p0 + p1);
    out[u * 2] = p0 * inv; out[u * 2 + 1] = p1 * inv;
}

static inline void launch_gemm(const float* X, const float* W, const float* bias, float* Y,
                               int M, int N, int K, int relu, hipStream_t s) {
    int tiles = (M / 16) * (N / 16);
    dim3 blk(32, 8);
    int grid = (tiles + 7) / 8;
    wmma_gemm_bias_act<<<grid, blk, 0, s>>>(X, W, bias, Y, M, N, K, relu);
}

extern "C" void kernel_launch(void* const* d_in, const int* in_sizes, int n_in,
                              void* d_out, int out_size, void* d_ws, size_t ws_size,
                              hipStream_t stream) {
    const float* J        = (const float*)d_in[0];
    const float* b        = (const float*)d_in[1];
    const int*   node_var = (const int*)d_in[2];
    const int*   e1_row   = (const int*)d_in[3];
    const int*   e1_col   = (const int*)d_in[4];
    const float* v2f_W1 = (const float*)d_in[5];  const float* v2f_b1 = (const float*)d_in[6];
    const float* v2f_W2 = (const float*)d_in[7];  const float* v2f_b2 = (const float*)d_in[8];
    const float* v2f_W3 = (const float*)d_in[9];  const float* v2f_b3 = (const float*)d_in[10];
    const float* f2v_W1 = (const float*)d_in[11]; const float* f2v_b1 = (const float*)d_in[12];
    const float* f2v_W2 = (const float*)d_in[13]; const float* f2v_b2 = (const float*)d_in[14];
    const float* f2v_W3 = (const float*)d_in[15]; const float* f2v_b3 = (const float*)d_in[16];
    const float* gv_Wih = (const float*)d_in[17]; const float* gv_Whh = (const float*)d_in[18];
    const float* gv_bih = (const float*)d_in[19]; const float* gv_bhh = (const float*)d_in[20];
    const float* gf_Wih = (const float*)d_in[21]; const float* gf_Whh = (const float*)d_in[22];
    const float* gf_bih = (const float*)d_in[23]; const float* gf_bhh = (const float*)d_in[24];
    const float* ro_W1  = (const float*)d_in[25]; const float* ro_b1  = (const float*)d_in[26];
    const float* ro_W2  = (const float*)d_in[27]; const float* ro_b2  = (const float*)d_in[28];
    const float* ro_W3  = (const float*)d_in[29]; const float* ro_b3  = (const float*)d_in[30];

    char* ws = (char*)d_ws;
    size_t off = 0;
    auto alloc = [&](size_t bytes) -> void* {
        void* p = ws + off;
        off = (off + bytes + 255) & ~(size_t)255;
        return p;
    };
    float* hv_a   = (float*)alloc((size_t)NMSG * SDIM * 4);
    float* hf_a   = (float*)alloc((size_t)NMSG * SDIM * 4);
    float* hv_b   = (float*)alloc((size_t)NMSG * SDIM * 4);
    float* hf_b   = (float*)alloc((size_t)NMSG * SDIM * 4);
    float* rawbuf = (float*)alloc((size_t)CHUNK * KPAD * 4);   // also reused as H2 [CHUNK,128]
    float* h1buf  = (float*)alloc((size_t)CHUNK * 128 * 4);
    float* em1    = (float*)alloc((size_t)E1_TOT * 2 * 4);
    float* em2    = (float*)alloc((size_t)NMSG * 2 * 4);
    float* nmbuf  = (float*)alloc((size_t)NMSG * 2 * 4);
    float* newbuf = (float*)alloc((size_t)NMSG * 2 * 4);
    float* w1p_v  = (float*)alloc((size_t)128 * KPAD * 4);
    float* w1p_f  = (float*)alloc((size_t)128 * KPAD * 4);
    float* nodemsg= (float*)alloc((size_t)N_VARS * SDIM * 4);
    float* l1buf  = (float*)alloc((size_t)N_VARS * 128 * 4);
    float* l2buf  = (float*)alloc((size_t)N_VARS * 128 * 4);
    int*   order  = (int*)alloc((size_t)N_VARS * 8 * 4);
    int*   invc   = (int*)alloc((size_t)NMSG * 4);

    hipMemsetAsync(hv_a, 0, (size_t)NMSG * SDIM * 4, stream);
    hipMemsetAsync(hf_a, 0, (size_t)NMSG * SDIM * 4, stream);

    pad_w1<<<(128 * KPAD + 255) / 256, 256, 0, stream>>>(v2f_W1, f2v_W1, w1p_v, w1p_f);
    build_order<<<(N_VARS + 255) / 256, 256, 0, stream>>>(node_var, order, invc);

    float* hv_c = hv_a; float* hv_n = hv_b;
    float* hf_c = hf_a; float* hf_n = hf_b;

    for (int step = 0; step < 10; ++step) {
        // -------- var -> fac --------
        for (int ch = 0; ch < NCHUNK; ++ch) {
            int ebase = ch * CHUNK;
            gather_v2f_raw<<<(CHUNK * KPAD + 255) / 256, 256, 0, stream>>>(
                rawbuf, hf_c, hv_c, e1_row, e1_col, node_var, b, ebase);
            launch_gemm(rawbuf, w1p_v, v2f_b1, h1buf, CHUNK, 128, KPAD, 1, stream);
            launch_gemm(h1buf, v2f_W2, v2f_b2, rawbuf, CHUNK, 128, 128, 1, stream);
            em_head<<<(CHUNK + 255) / 256, 256, 0, stream>>>(
                rawbuf, v2f_W3, v2f_b3, em1, CHUNK, ebase);
        }
        nm_gather<<<(NMSG + 255) / 256, 256, 0, stream>>>(em1, node_var, invc, nmbuf);
        gru_update<<<(NMSG * SDIM + 255) / 256, 256, 0, stream>>>(
            nmbuf, hv_c, hv_n, gv_Wih, gv_Whh, gv_bih, gv_bhh, NMSG, 0);
        // -------- fac -> var --------
        gather_f2v_raw<<<(NMSG * KPAD + 255) / 256, 256, 0, stream>>>(
            rawbuf, hv_n, hf_c, node_var, b);
        launch_gemm(rawbuf, w1p_f, f2v_b1, h1buf, NMSG, 128, KPAD, 1, stream);
        launch_gemm(h1buf, f2v_W2, f2v_b2, rawbuf, NMSG, 128, 128, 1, stream);
        em_head<<<(NMSG + 255) / 256, 256, 0, stream>>>(
            rawbuf, f2v_W3, f2v_b3, em2, NMSG, 0);
        lse_factor<<<(NMSG + 255) / 256, 256, 0, stream>>>(em2, J, node_var, newbuf);
        gru_update<<<(NMSG * SDIM + 255) / 256, 256, 0, stream>>>(
            newbuf, hf_c, hf_n, gf_Wih, gf_Whh, gf_bih, gf_bhh, NMSG, 1);
        // ping-pong
        float* t;
        t = hv_c; hv_c = hv_n; hv_n = t;
        t = hf_c; hf_c = hf_n; hf_n = t;
    }

    // -------- readout --------
    node_msg_sum<<<(N_VARS * SDIM + 255) / 256, 256, 0, stream>>>(hf_c, order, nodemsg);
    launch_gemm(nodemsg, ro_W1, ro_b1, l1buf, N_VARS, 128, 64, 1, stream);
    launch_gemm(l1buf, ro_W2, ro_b2, l2buf, N_VARS, 128, 128, 1, stream);
    readout_softmax<<<(N_VARS + 255) / 256, 256, 0, stream>>>(l2buf, ro_W3, ro_b3, (float*)d_out);
}